// VirtSAGEMolNode_64355789963804
// MI455X (gfx1250) — compile-verified
//
#include <hip/hip_runtime.h>

// VirtSAGEMolNode forward for MI455X (gfx1250, wave32, WMMA).
// - Graph gather/scatter phases are L2-resident (h/agg = 51MB each, L2 = 192MB)
//   and bandwidth bound -> fp32 WMMA 16x16x4 for the GEMMs (exact fp32; matrix
//   throughput is nowhere near the bottleneck, so no reason to drop precision).
// - Weights are repacked once per launch into WMMA-fragment-major layout so
//   each v_wmma step consumes one coalesced global_load_b64 per lane.

#define NN 100000
#define EE 600000
#define GG 512
#define DD 128
#define LL 5
#define AVV 120
#define BVV 8
#define EPSF 1e-5f
#define LDA 260   // padded LDS row stride (floats): 260 = 4*65, conflict-free

typedef __attribute__((ext_vector_type(2))) float v2f;
typedef __attribute__((ext_vector_type(8))) float v8f;

// ---- WMMA inner strip ------------------------------------------------------
// One wave computes a 16x16 fp32 tile: C += A(16xK, LDS) * B (packed global).
// A-frag (ISA 7.12.2, 32-bit A 16x4): lane&15 = M row, lane>>4 selects K pair.
// B is pre-packed so lane's two values are contiguous: 256B/wave per step.
__device__ __forceinline__ v8f wmma_strip(const float* A, int kOff, int Kquads,
                                          const float* __restrict__ P,
                                          int ct, int lane, v8f c) {
  const int m  = lane & 15;
  const int kh = (lane >> 4) * 2;                   // 0 or 2
  const float* arow = A + m * LDA + kOff + kh;
  const float* pb   = P + ((size_t)(ct * Kquads) * 32 + lane) * 2;
  for (int kq = 0; kq < Kquads; ++kq) {
    v2f a = *(const v2f*)(arow + kq * 4);           // ds_read_b64
    v2f b = *(const v2f*)(pb + kq * 64);            // global_load_b64, coalesced
    c = __builtin_amdgcn_wmma_f32_16x16x4_f32(false, a, false, b,
                                              (short)0, c, false, false);
  }
  return c;
}

// Repack W[K x Dout] (row-major) into fragment-major:
// P[((ct*Kq + kq)*32 + lane)*2 + {0,1}] = W[kq*4 + 2*(lane>>4) + {0,1}][ct*16 + (lane&15)]
__global__ void k_pack(const float* __restrict__ W, float* __restrict__ P,
                       int Kquads, int Dout) {
  int idx  = blockIdx.x * blockDim.x + threadIdx.x;  // (Dout/16)*Kquads*32
  int lane = idx & 31;
  int t    = idx >> 5;
  int kq   = t % Kquads;
  int ct   = t / Kquads;
  int col  = ct * 16 + (lane & 15);
  int k0   = kq * 4 + (lane >> 4) * 2;
  float2 v = { W[(size_t)k0 * Dout + col], W[(size_t)(k0 + 1) * Dout + col] };
  *(float2*)&P[(size_t)idx * 2] = v;
}

// ---- graph / elementwise kernels -------------------------------------------

__global__ void k_count(const int* __restrict__ ei, float* __restrict__ cnt) {
  int e = blockIdx.x * blockDim.x + threadIdx.x;
  if (e < EE) atomicAdd(&cnt[ei[EE + e]], 1.0f);
}

__global__ void k_dinv(float* __restrict__ d) {
  int n = blockIdx.x * blockDim.x + threadIdx.x;
  if (n < NN) d[n] = 1.0f / fmaxf(d[n], 1.0f);
}

__global__ void k_atom(const int* __restrict__ x, const float* __restrict__ emb,
                       float* __restrict__ h) {
  int idx = blockIdx.x * blockDim.x + threadIdx.x;   // exactly N*D threads
  int n = idx >> 7, j = idx & 127;
  float s = 0.f;
#pragma unroll
  for (int f = 0; f < 9; ++f)
    s += emb[((size_t)f * AVV + x[n * 9 + f]) * DD + j];
  h[idx] = s;
}

__global__ void k_init_virt(const float* __restrict__ ve, float* __restrict__ v) {
  int idx = blockIdx.x * blockDim.x + threadIdx.x;
  if (idx < GG * DD) v[idx] = ve[idx & 127];
}

__global__ void k_add_virt_pool(float* __restrict__ h, const float* __restrict__ virt,
                                const int* __restrict__ batch,
                                float* __restrict__ pooled, int do_pool) {
  int idx = blockIdx.x * blockDim.x + threadIdx.x;   // N*D threads
  int n = idx >> 7, j = idx & 127;
  int g = batch[n];
  float v = h[idx] + virt[g * DD + j];
  h[idx] = v;
  if (do_pool) atomicAdd(&pooled[g * DD + j], v);
}

// One wave32 per edge, float4 per lane (32*4 = 128 channels); prefetch the
// next edge-group's h[src] row to hide L2 latency on the random gather.
__global__ __launch_bounds__(256) void k_edge(const float* __restrict__ h,
                                              const int* __restrict__ ei,
                                              const int* __restrict__ ea,
                                              const float* __restrict__ bond,
                                              float* __restrict__ agg) {
  int e    = blockIdx.x * 8 + (threadIdx.x >> 5);    // E % 8 == 0
  int lane = threadIdx.x & 31;
  int c    = lane * 4;
  int src = ei[e], dst = ei[EE + e];
  if (e + 8 < EE) {
    int srcn = ei[e + 8];
    __builtin_prefetch(&h[(size_t)srcn * DD + c], 0, 3);  // global_prefetch_b8
  }
  int a0 = ea[e * 3 + 0], a1 = ea[e * 3 + 1], a2 = ea[e * 3 + 2];
  float4 hv = *(const float4*)&h[(size_t)src * DD + c];
  float4 e0 = *(const float4*)&bond[((size_t)0 * BVV + a0) * DD + c];
  float4 e1 = *(const float4*)&bond[((size_t)1 * BVV + a1) * DD + c];
  float4 e2 = *(const float4*)&bond[((size_t)2 * BVV + a2) * DD + c];
  float m0 = fmaxf(hv.x + e0.x + e1.x + e2.x, 0.f);
  float m1 = fmaxf(hv.y + e0.y + e1.y + e2.y, 0.f);
  float m2 = fmaxf(hv.z + e0.z + e1.z + e2.z, 0.f);
  float m3 = fmaxf(hv.w + e0.w + e1.w + e2.w, 0.f);
  float* o = &agg[(size_t)dst * DD + c];
  atomicAdd(o + 0, m0); atomicAdd(o + 1, m1);
  atomicAdd(o + 2, m2); atomicAdd(o + 3, m3);
}

// out = [agg*dinv | h] @ [Wl;Wr] + bl, in place over agg (each block stages its
// own 16 rows into LDS before storing). Accumulates column sum/sumsq for BN.
__global__ __launch_bounds__(256) void k_gemm_node(
    const float* agg, const float* __restrict__ dinv, const float* __restrict__ h,
    const float* __restrict__ pWl, const float* __restrict__ bl,
    const float* __restrict__ pWr, float* out,
    float* __restrict__ colsum, float* __restrict__ colsq) {
  __shared__ float A[16 * LDA];
  int n0 = blockIdx.x * 16;                          // N % 16 == 0
  for (int i = threadIdx.x; i < 16 * 256; i += 256) {
    int r = i >> 8, k = i & 255;
    int n = n0 + r;
    float v = (k < 128) ? agg[(size_t)n * DD + k] * dinv[n]
                        : h[(size_t)n * DD + (k - 128)];
    A[r * LDA + k] = v;
  }
  __syncthreads();
  int lane = threadIdx.x & 31;
  int ct   = threadIdx.x >> 5;                       // 8 waves -> 8 col tiles
  v8f c = {};
  c = wmma_strip(A, 0,   32, pWl, ct, lane, c);      // K = 128 (32 quads)
  c = wmma_strip(A, 128, 32, pWr, ct, lane, c);
  int m0  = (lane >> 4) * 8;
  int col = ct * 16 + (lane & 15);
  float bv = bl[col];
  float s = 0.f, s2 = 0.f;
#pragma unroll
  for (int g = 0; g < 8; ++g) {
    float v = c[g] + bv;
    out[(size_t)(n0 + m0 + g) * DD + col] = v;
    s += v; s2 += v * v;
  }
  atomicAdd(&colsum[col], s);
  atomicAdd(&colsq[col], s2);
}

// Generic small GEMM for the virtual-node MLP: out = (A [+ addv]) @ W + bias.
__global__ __launch_bounds__(256) void k_gemm_mlp(
    const float* __restrict__ Ain, const float* __restrict__ addv,
    const float* __restrict__ P, const float* __restrict__ bias,
    float* __restrict__ out, int Din, int Dout,
    float* __restrict__ colsum, float* __restrict__ colsq) {
  __shared__ float A[16 * LDA];
  int n0 = blockIdx.x * 16;
  for (int i = threadIdx.x; i < 16 * Din; i += 256) {
    int r = i / Din, k = i - r * Din;
    int n = n0 + r;
    float v = Ain[(size_t)n * Din + k];
    if (addv) v += addv[(size_t)n * Din + k];
    A[r * LDA + k] = v;
  }
  __syncthreads();
  int lane = threadIdx.x & 31;
  int ct   = blockIdx.y * 8 + (threadIdx.x >> 5);
  v8f c = {};
  c = wmma_strip(A, 0, Din >> 2, P, ct, lane, c);
  int m0  = (lane >> 4) * 8;
  int col = ct * 16 + (lane & 15);
  float bv = bias[col];
  float s = 0.f, s2 = 0.f;
#pragma unroll
  for (int g = 0; g < 8; ++g) {
    float v = c[g] + bv;
    out[(size_t)(n0 + m0 + g) * Dout + col] = v;
    s += v; s2 += v * v;
  }
  atomicAdd(&colsum[col], s);
  atomicAdd(&colsq[col], s2);
}

// Fold BN stats into per-column scale/shift (in place over colsum/colsq).
__global__ void k_bn_prep(float* __restrict__ colsum, float* __restrict__ colsq,
                          const float* __restrict__ g, const float* __restrict__ b,
                          float invM) {
  int j = threadIdx.x;                               // one block of C threads
  float mean = colsum[j] * invM;
  float var  = colsq[j] * invM - mean * mean;
  float sc   = g[j] * rsqrtf(var + EPSF);
  colsum[j] = sc;
  colsq[j]  = b[j] - mean * sc;
}

// Apply folded BN (+ optional ReLU): out = in*scale[j] + shift[j].
__global__ void k_bn_apply(const float* __restrict__ in,
                           const float* __restrict__ scale,
                           const float* __restrict__ shift,
                           float* __restrict__ out, int relu, int C) {
  int idx = blockIdx.x * blockDim.x + threadIdx.x;
  int j = idx & (C - 1);                             // C is 128 or 256
  float v = fmaf(in[idx], scale[j], shift[j]);
  if (relu) v = fmaxf(v, 0.f);
  out[idx] = v;
}

// ---- host orchestration ----------------------------------------------------

extern "C" void kernel_launch(void* const* d_in, const int* in_sizes, int n_in,
                              void* d_out, int out_size, void* d_ws, size_t ws_size,
                              hipStream_t stream) {
  (void)in_sizes; (void)n_in; (void)out_size; (void)ws_size;
  const int*   x     = (const int*)d_in[0];
  const int*   ei    = (const int*)d_in[1];
  const int*   ea    = (const int*)d_in[2];
  const int*   batch = (const int*)d_in[3];
  const float* atomE = (const float*)d_in[4];
  const float* bondE = (const float*)d_in[5];
  const float* llw   = (const float*)d_in[6];
  const float* llb   = (const float*)d_in[7];
  const float* lrw   = (const float*)d_in[8];
  const float* bng   = (const float*)d_in[9];
  const float* bnb   = (const float*)d_in[10];
  const float* vemb  = (const float*)d_in[11];
  const float* w1    = (const float*)d_in[12];
  const float* b1    = (const float*)d_in[13];
  const float* bn1g  = (const float*)d_in[14];
  const float* bn1b  = (const float*)d_in[15];
  const float* w2    = (const float*)d_in[16];
  const float* b2    = (const float*)d_in[17];
  const float* bn2g  = (const float*)d_in[18];
  const float* bn2b  = (const float*)d_in[19];

  float* p = (float*)d_ws;
  float* h      = p; p += (size_t)NN * DD;
  float* h2     = p; p += (size_t)NN * DD;
  float* agg    = p; p += (size_t)NN * DD;     // also serves as GEMM output
  float* dinv   = p; p += NN;
  float* pooled = p; p += GG * DD;
  float* z      = p; p += GG * 2 * DD;
  float* virt   = p; p += GG * DD;
  float* virt2  = p; p += GG * DD;
  float* colsum = p; p += 2 * DD;
  float* colsq  = p; p += 2 * DD;
  float* pWl    = p; p += (size_t)LL * DD * DD;
  float* pWr    = p; p += (size_t)LL * DD * DD;
  float* pW1    = p; p += (size_t)(LL - 1) * DD * 2 * DD;
  float* pW2    = p; p += (size_t)(LL - 1) * 2 * DD * DD;

  const float invN = 1.0f / (float)NN;
  const float invG = 1.0f / (float)GG;

  // Repack all weights into WMMA-fragment-major layout (tiny; once per launch).
  for (int l = 0; l < LL; ++l) {
    k_pack<<<DD * DD / 2 / 256, 256, 0, stream>>>(
        llw + (size_t)l * DD * DD, pWl + (size_t)l * DD * DD, DD / 4, DD);
    k_pack<<<DD * DD / 2 / 256, 256, 0, stream>>>(
        lrw + (size_t)l * DD * DD, pWr + (size_t)l * DD * DD, DD / 4, DD);
  }
  for (int l = 0; l < LL - 1; ++l) {
    k_pack<<<DD * 2 * DD / 2 / 256, 256, 0, stream>>>(
        w1 + (size_t)l * DD * 2 * DD, pW1 + (size_t)l * DD * 2 * DD, DD / 4, 2 * DD);
    k_pack<<<2 * DD * DD / 2 / 256, 256, 0, stream>>>(
        w2 + (size_t)l * 2 * DD * DD, pW2 + (size_t)l * 2 * DD * DD, 2 * DD / 4, DD);
  }

  hipMemsetAsync(dinv, 0, NN * sizeof(float), stream);
  k_count<<<(EE + 255) / 256, 256, 0, stream>>>(ei, dinv);
  k_dinv<<<(NN + 255) / 256, 256, 0, stream>>>(dinv);
  k_atom<<<NN * DD / 256, 256, 0, stream>>>(x, atomE, h);
  k_init_virt<<<GG * DD / 256, 256, 0, stream>>>(vemb, virt);

  for (int l = 0; l < LL; ++l) {
    const int last = (l == LL - 1);
    hipMemsetAsync(agg, 0, (size_t)NN * DD * sizeof(float), stream);
    hipMemsetAsync(colsum, 0, 2 * DD * sizeof(float), stream);
    hipMemsetAsync(colsq, 0, 2 * DD * sizeof(float), stream);
    if (!last) hipMemsetAsync(pooled, 0, GG * DD * sizeof(float), stream);

    k_add_virt_pool<<<NN * DD / 256, 256, 0, stream>>>(h, virt, batch, pooled, !last);
    k_edge<<<EE / 8, 256, 0, stream>>>(h, ei, ea,
        bondE + (size_t)l * 3 * BVV * DD, agg);
    k_gemm_node<<<NN / 16, 256, 0, stream>>>(agg, dinv, h,
        pWl + (size_t)l * DD * DD, llb + l * DD, pWr + (size_t)l * DD * DD,
        agg, colsum, colsq);
    k_bn_prep<<<1, DD, 0, stream>>>(colsum, colsq, bng + l * DD, bnb + l * DD, invN);
    k_bn_apply<<<NN * DD / 256, 256, 0, stream>>>(agg, colsum, colsq, h2, !last, DD);

    if (!last) {
      hipMemsetAsync(colsum, 0, 2 * DD * sizeof(float), stream);
      hipMemsetAsync(colsq, 0, 2 * DD * sizeof(float), stream);
      k_gemm_mlp<<<dim3(GG / 16, 2), 256, 0, stream>>>(pooled, virt,
          pW1 + (size_t)l * DD * 2 * DD, b1 + l * 2 * DD, z, DD, 2 * DD,
          colsum, colsq);
      k_bn_prep<<<1, 2 * DD, 0, stream>>>(colsum, colsq, bn1g + l * 2 * DD,
                                          bn1b + l * 2 * DD, invG);
      k_bn_apply<<<GG * 2 * DD / 256, 256, 0, stream>>>(z, colsum, colsq, z, 1, 2 * DD);

      hipMemsetAsync(colsum, 0, 2 * DD * sizeof(float), stream);
      hipMemsetAsync(colsq, 0, 2 * DD * sizeof(float), stream);
      k_gemm_mlp<<<dim3(GG / 16, 1), 256, 0, stream>>>(z, nullptr,
          pW2 + (size_t)l * 2 * DD * DD, b2 + l * DD, virt2, 2 * DD, DD,
          colsum, colsq);
      k_bn_prep<<<1, DD, 0, stream>>>(colsum, colsq, bn2g + l * DD,
                                      bn2b + l * DD, invG);
      k_bn_apply<<<GG * DD / 256, 256, 0, stream>>>(virt2, colsum, colsq,
                                                    virt2, 1, DD);
      float* t = virt; virt = virt2; virt2 = t;
    }
    float* t = h; h = h2; h2 = t;
  }

  hipMemcpyAsync(d_out, h, (size_t)NN * DD * sizeof(float),
                 hipMemcpyDeviceToDevice, stream);
}